// CrossCompress_14542759264545
// MI455X (gfx1250) — compile-verified
//
#include <hip/hip_runtime.h>
#include <stdint.h>

typedef __attribute__((ext_vector_type(2))) float v2f;
typedef __attribute__((ext_vector_type(4))) float v4f;
typedef __attribute__((ext_vector_type(8))) float v8f;

#define DIMSZ  128
#define PITCH  132   // tile row pitch (dwords): stride-4-bank pattern -> conflict-free WMMA A reads
#define WPITCH 136   // weight table pitch (dwords): stride-8-bank pattern -> conflict-free B reads
#define WAVES_PER_BLOCK 2
#define ROWS_PER_WAVE   16

__device__ __forceinline__ void async_load_b128(uint32_t lds_byte, const float* gaddr) {
    // GLOBAL_LOAD_ASYNC_TO_LDS_B128: per-lane LDS byte address in VGPR, 64-bit global addr.
    // Tracked by ASYNCcnt (CDNA5 ISA).
    asm volatile("global_load_async_to_lds_b128 %0, %1, off"
                 :: "v"(lds_byte), "v"(gaddr) : "memory");
}

__global__ __launch_bounds__(WAVES_PER_BLOCK * 32)
void cross_compress_kernel(const float* __restrict__ v,
                           const float* __restrict__ e,
                           const float* __restrict__ w_vv,
                           const float* __restrict__ w_ev,
                           const float* __restrict__ w_ve,
                           const float* __restrict__ w_ee,
                           const float* __restrict__ bias_v,
                           const float* __restrict__ bias_e,
                           float* __restrict__ out_v,
                           float* __restrict__ out_e,
                           int nrows) {
    __shared__ float vtile[WAVES_PER_BLOCK][ROWS_PER_WAVE][PITCH];
    __shared__ float etile[WAVES_PER_BLOCK][ROWS_PER_WAVE][PITCH];
    __shared__ float wtabE[3][WPITCH];   // rows: {w_vv, w_ve, zeros}
    __shared__ float wtabV[3][WPITCH];   // rows: {w_ev, w_ee, zeros}
    __shared__ float dots[WAVES_PER_BLOCK][ROWS_PER_WAVE][4]; // {s_vv, s_ev, s_ve, s_ee}

    const int tid  = threadIdx.x;
    const int wave = tid >> 5;
    const int lane = tid & 31;
    const int row0 = (blockIdx.x * WAVES_PER_BLOCK + wave) * ROWS_PER_WAVE; // wave-uniform
    const bool active = row0 < nrows;

    // ---- Phase 0: fire-and-forget async tile loads (16 x b128 per tensor per wave) ----
    if (active) {
        const float* gv = v + (size_t)row0 * DIMSZ + lane * 4;
        const float* ge = e + (size_t)row0 * DIMSZ + lane * 4;
        const uint32_t lv = (uint32_t)(uintptr_t)&vtile[wave][0][lane * 4];
        const uint32_t le = (uint32_t)(uintptr_t)&etile[wave][0][lane * 4];
        #pragma unroll
        for (int i = 0; i < ROWS_PER_WAVE; ++i) {
            async_load_b128(lv + i * (PITCH * 4), gv + i * DIMSZ);
            async_load_b128(le + i * (PITCH * 4), ge + i * DIMSZ);
        }
    }

    // ---- Phase 1: build branch-free weight tables (overlaps with async DMA) ----
    for (int k = tid; k < DIMSZ; k += WAVES_PER_BLOCK * 32) {
        wtabE[0][k] = w_vv[k];  wtabE[1][k] = w_ve[k];  wtabE[2][k] = 0.f;
        wtabV[0][k] = w_ev[k];  wtabV[1][k] = w_ee[k];  wtabV[2][k] = 0.f;
    }
    __syncthreads();

    // ---- Phase 2: fp32 WMMA dot products, all operands from LDS ----
    if (active) {
        asm volatile("s_wait_asynccnt 0x0" ::: "memory");

        const int n     = lane & 15;          // A: M-row, B/C: N-col
        const int khalf = (lane >> 4) * 2;    // K sub-offset {0,2} within 4-wide step
        const int wrow  = (n < 2) ? n : 2;    // branchless B-column select (row 2 = zeros)

        v8f accE = {0.f,0.f,0.f,0.f,0.f,0.f,0.f,0.f};  // E x [w_vv | w_ve]
        v8f accV = {0.f,0.f,0.f,0.f,0.f,0.f,0.f,0.f};  // V x [w_ev | w_ee]

        #pragma unroll 8
        for (int kk = 0; kk < DIMSZ / 4; ++kk) {
            const int kb = kk * 4 + khalf;
            v2f aE = *(const v2f*)&etile[wave][n][kb];
            v2f aV = *(const v2f*)&vtile[wave][n][kb];
            v2f bE = *(const v2f*)&wtabE[wrow][kb];
            v2f bV = *(const v2f*)&wtabV[wrow][kb];
            accE = __builtin_amdgcn_wmma_f32_16x16x4_f32(
                false, aE, false, bE, (short)0, accE, false, false);
            accV = __builtin_amdgcn_wmma_f32_16x16x4_f32(
                false, aV, false, bV, (short)0, accV, false, false);
        }

        // C-fragment layout: VGPR r, lane l -> D[M = r + 8*(l>=16)][N = l&15]
        const int rbase = (lane >> 4) * 8;
        if (n < 2) {
            #pragma unroll
            for (int r = 0; r < 8; ++r) {
                const int rr = rbase + r;
                if (n == 0) {
                    dots[wave][rr][0] = accE[r];   // s_vv = e . w_vv
                    dots[wave][rr][1] = accV[r];   // s_ev = v . w_ev
                } else {
                    dots[wave][rr][2] = accE[r];   // s_ve = e . w_ve
                    dots[wave][rr][3] = accV[r];   // s_ee = v . w_ee
                }
            }
        }
    }
    __syncthreads();
    if (!active) return;

    // ---- Phase 3: elementwise scale+bias from LDS, coalesced B128 stores ----
    const int c0 = lane * 4;
    v4f bv = *(const v4f*)(bias_v + c0);
    v4f be = *(const v4f*)(bias_e + c0);

    #pragma unroll 4
    for (int i = 0; i < ROWS_PER_WAVE; ++i) {
        v4f vv = *(const v4f*)&vtile[wave][i][c0];
        v4f ee = *(const v4f*)&etile[wave][i][c0];
        const float s_vv = dots[wave][i][0];
        const float s_ev = dots[wave][i][1];
        const float s_ve = dots[wave][i][2];
        const float s_ee = dots[wave][i][3];
        v4f ov = vv * s_vv + ee * s_ev + bv;
        v4f oe = vv * s_ve + ee * s_ee + be;
        const size_t off = (size_t)(row0 + i) * DIMSZ + c0;
        *(v4f*)(out_v + off) = ov;
        *(v4f*)(out_e + off) = oe;
    }
}

extern "C" void kernel_launch(void* const* d_in, const int* in_sizes, int n_in,
                              void* d_out, int out_size, void* d_ws, size_t ws_size,
                              hipStream_t stream) {
    const float* v   = (const float*)d_in[0];
    const float* e   = (const float*)d_in[1];
    const float* wvv = (const float*)d_in[2];
    const float* wev = (const float*)d_in[3];
    const float* wve = (const float*)d_in[4];
    const float* wee = (const float*)d_in[5];
    const float* bv  = (const float*)d_in[6];
    const float* be  = (const float*)d_in[7];

    const int nrows = in_sizes[0] / DIMSZ;           // 16384
    float* out_v = (float*)d_out;
    float* out_e = out_v + (size_t)nrows * DIMSZ;

    const int tiles  = (nrows + ROWS_PER_WAVE - 1) / ROWS_PER_WAVE;
    const int blocks = (tiles + WAVES_PER_BLOCK - 1) / WAVES_PER_BLOCK;

    hipLaunchKernelGGL(cross_compress_kernel,
                       dim3(blocks), dim3(WAVES_PER_BLOCK * 32), 0, stream,
                       v, e, wvv, wev, wve, wee, bv, be, out_v, out_e, nrows);
}